// DecoderRNN_40123584479689
// MI455X (gfx1250) — compile-verified
//
#include <hip/hip_runtime.h>
#include <math.h>

// Problem constants (from the reference)
#define BB 128
#define LL 196
#define DD 512
#define TT 20
#define VV 10000
#define EE 256
#define HH 512
#define AA 512
#define G4 2048  // 4*H

typedef float v2f __attribute__((ext_vector_type(2)));
typedef float v8f __attribute__((ext_vector_type(8)));

// ---------------------------------------------------------------------------
// CDNA5 async global->LDS path (ASYNCcnt-tracked), with safe fallback.
// Probe result (round 2): builtin exists; param type is v4i* (16B vector),
// global pointer in addrspace(1), LDS pointer in addrspace(3).
// ---------------------------------------------------------------------------
#if defined(__gfx1250__) && defined(__has_builtin)
#if __has_builtin(__builtin_amdgcn_global_load_async_to_lds_b128) && \
    __has_builtin(__builtin_amdgcn_s_wait_asynccnt)
#define HAS_ASYNC_LDS 1
#endif
#endif

#ifdef HAS_ASYNC_LDS
typedef int v4i_a __attribute__((ext_vector_type(4)));
typedef __attribute__((address_space(1))) v4i_a gv4i;
typedef __attribute__((address_space(3))) v4i_a sv4i;
#define ASYNC_F4(gp, lp)                                                    \
    __builtin_amdgcn_global_load_async_to_lds_b128((gv4i*)(gp),             \
                                                   (sv4i*)(lp), 0, 0)
#define ASYNC_WAIT() __builtin_amdgcn_s_wait_asynccnt(0)
#else
#define ASYNC_F4(gp, lp) (*(float4*)(lp) = *(const float4*)(gp))
#define ASYNC_WAIT() ((void)0)
#endif

// ---------------------------------------------------------------------------
// fp32 WMMA GEMM:  C[M,N] = act( beta*C + A[M,K] @ B[K,N] + bias ), row-major.
// Requirements here: K % 16 == 0 (all call sites: 256/512). M,N arbitrary
// (OOB B columns zero-filled so WMMA runs with EXEC all-ones; stores guarded).
//
// 256 threads = 8 waves; block tile 64x64; wave grid 2(M) x 4(N); each wave
// owns two 16x16 V_WMMA_F32_16X16X4_F32 tiles (32x16) sharing one B fragment.
// K-chunks of 16 are double-buffered in LDS via async global->LDS b128 loads.
//
// LDS pitches: A rows padded to 20 dwords, B rows to 72 dwords -> the strided
// fragment reads (row-stride 20, resp. +2-column halves) hit distinct banks
// for all 32 lanes, and every b128 transfer stays 16-byte aligned.
//
// Fragment layout per CDNA5 ISA 7.12.2 (wave32):
//   A 16x4 : lanes 0-15 -> M=lane, (K,K+1) in (v0,v1); lanes 16-31 -> (K+2,K+3)
//   B 4x16 : lanes 0-15 -> N=lane, rows K/K+1;          lanes 16-31 -> K+2/K+3
//   C 16x16: vgpr r -> M = r + 8*(lane>=16), N = lane&15
// ---------------------------------------------------------------------------
#define KC 16

__global__ __launch_bounds__(256) void gemm_wmma_f32(
    const float* __restrict__ A, const float* __restrict__ B,
    float* __restrict__ C, const float* __restrict__ bias,
    int M, int N, int K, int lda, int ldb, int ldc,
    int accumulate, int act_tanh)
{
    __shared__ float As[2][64][20];
    __shared__ float Bs[2][16][72];

    const int tid  = threadIdx.x;
    const int lane = tid & 31;
    const int wave = tid >> 5;
    const int bm = blockIdx.y;
    const int bn = blockIdx.x;

    // cooperative staging indices (one float4 per thread per tile)
    const int ar = tid >> 2;            // A row within tile   0..63
    const int ac = (tid & 3) * 4;       // A col within chunk  0,4,8,12
    const int br = tid >> 4;            // B row within chunk  0..15
    const int bc = (tid & 15) * 4;      // B col within tile   0..60

    const int  arow = bm * 64 + ar;
    const int  bcol = bn * 64 + bc;
    const bool a_ok = (arow < M);
    const bool b_ok = (bcol < N);

    const float* Ag = A + (long)arow * lda + ac;
    const float* Bg = B + (long)br * ldb + bcol;

    auto stage = [&](int c, int buf) {
        const int k0 = c * KC;
        if (a_ok) {
            ASYNC_F4(Ag + k0, &As[buf][ar][ac]);
        } else {
            float4 z = {0.f, 0.f, 0.f, 0.f};
            *(float4*)&As[buf][ar][ac] = z;
        }
        if (b_ok) {
            ASYNC_F4(Bg + (long)k0 * ldb, &Bs[buf][br][bc]);
        } else {
            float4 z = {0.f, 0.f, 0.f, 0.f};
            *(float4*)&Bs[buf][br][bc] = z;
        }
    };

    // wave tile position
    const int wm   = (wave >> 2) * 32;  // 0 or 32
    const int wn   = (wave & 3) * 16;   // 0..48
    const int half = lane >> 4;
    const int mr   = lane & 15;

    v8f acc0 = {0.f, 0.f, 0.f, 0.f, 0.f, 0.f, 0.f, 0.f};
    v8f acc1 = {0.f, 0.f, 0.f, 0.f, 0.f, 0.f, 0.f, 0.f};

    const int nChunks = K / KC;
    stage(0, 0);
    ASYNC_WAIT();
    __syncthreads();

    for (int c = 0; c < nChunks; ++c) {
        const int buf = c & 1;
        if (c + 1 < nChunks) {
            stage(c + 1, buf ^ 1);
#ifndef HAS_ASYNC_LDS
            // gfx1250 global_prefetch_b8 of the chunk after next
            if (c + 2 < nChunks) {
                __builtin_prefetch(Ag + (c + 2) * KC, 0, 1);
                __builtin_prefetch(Bg + (long)(c + 2) * KC * ldb, 0, 1);
            }
#endif
        }
#pragma unroll
        for (int kk = 0; kk < KC; kk += 4) {
            const int ka = kk + 2 * half;
            v2f a0, a1, bf;
            a0.x = As[buf][wm + mr][ka];
            a0.y = As[buf][wm + mr][ka + 1];
            a1.x = As[buf][wm + 16 + mr][ka];
            a1.y = As[buf][wm + 16 + mr][ka + 1];
            bf.x = Bs[buf][ka][wn + mr];
            bf.y = Bs[buf][ka + 1][wn + mr];
            acc0 = __builtin_amdgcn_wmma_f32_16x16x4_f32(
                false, a0, false, bf, (short)0, acc0, false, false);
            acc1 = __builtin_amdgcn_wmma_f32_16x16x4_f32(
                false, a1, false, bf, (short)0, acc1, false, false);
        }
        ASYNC_WAIT();
        __syncthreads();
    }

    // epilogue (per-lane guards are fine: no more WMMA after this point)
    const int ccol = bn * 64 + wn + mr;
    if (ccol < N) {
        const float bval = bias ? bias[ccol] : 0.f;
#pragma unroll
        for (int r = 0; r < 8; ++r) {
            const int row0 = bm * 64 + wm + r + 8 * half;
            const int row1 = row0 + 16;
            if (row0 < M) {
                float val = acc0[r] + bval;
                if (accumulate) val += C[(long)row0 * ldc + ccol];
                if (act_tanh) val = tanhf(val);
                C[(long)row0 * ldc + ccol] = val;
            }
            if (row1 < M) {
                float val = acc1[r] + bval;
                if (accumulate) val += C[(long)row1 * ldc + ccol];
                if (act_tanh) val = tanhf(val);
                C[(long)row1 * ldc + ccol] = val;
            }
        }
    }
}

// ---------------------------------------------------------------------------
// mean over L:  mean[b,d] = (1/L) * sum_l a[b,l,d]
// ---------------------------------------------------------------------------
__global__ __launch_bounds__(256) void mean_ctx_kernel(
    const float* __restrict__ a, float* __restrict__ mean)
{
    int idx = blockIdx.x * 256 + threadIdx.x;   // b*D + d
    if (idx >= BB * DD) return;
    int b = idx >> 9, d = idx & (DD - 1);
    const float* p = a + (long)b * LL * DD + d;
    float s = 0.f;
    for (int l = 0; l < LL; ++l) s += p[(long)l * DD];
    mean[idx] = s * (1.0f / LL);
}

// Whc = Wh + Wc (prev_h == h in the reference carry, so both attention
// projections collapse into one GEMM against Wh+Wc)
__global__ __launch_bounds__(256) void add_mat_kernel(
    const float* __restrict__ X, const float* __restrict__ Y,
    float* __restrict__ Z, int n)
{
    int i = blockIdx.x * 256 + threadIdx.x;
    if (i < n) Z[i] = X[i] + Y[i];
}

// embX[(t*B+b), e] = embed_W[captions[b,t], e]   for t in 0..T-2 (time-major)
__global__ __launch_bounds__(256) void embed_gather_kernel(
    const int* __restrict__ captions, const float* __restrict__ embed_W,
    float* __restrict__ embX)
{
    int idx = blockIdx.x * 256 + threadIdx.x;
    const int n = (TT - 1) * BB * EE;
    if (idx >= n) return;
    int row = idx / EE, e = idx - row * EE;
    int t = row / BB, b = row - t * BB;
    int tok = captions[b * TT + t];
    embX[idx] = embed_W[(long)tok * EE + e];
}

__device__ __forceinline__ float warpReduceSum(float v) {
    for (int o = 16; o > 0; o >>= 1) v += __shfl_down(v, o);
    return v;
}
__device__ __forceinline__ float warpReduceMax(float v) {
    for (int o = 16; o > 0; o >>= 1) v = fmaxf(v, __shfl_down(v, o));
    return v;
}

// ---------------------------------------------------------------------------
// Fused attention for one timestep. One block per batch row b (8 waves):
//   e[l]   = sum_k v[k] * tanh(wa[b,l,k] + hsum[b,k])
//   alpha  = softmax_L(e);  beta = sigmoid(h.beta_W + beta_b)
//   ctx[d] = beta * sum_l alpha[l] * a[b,l,d]
// ---------------------------------------------------------------------------
__global__ __launch_bounds__(256) void attention_kernel(
    const float* __restrict__ wa, const float* __restrict__ hsum,
    const float* __restrict__ vvec, const float* __restrict__ h,
    const float* __restrict__ betaW, const float* __restrict__ betab,
    const float* __restrict__ a, float* __restrict__ ctx,
    float* __restrict__ alphas, int t)
{
    __shared__ float s_h[AA];
    __shared__ float s_v[AA];
    __shared__ float s_e[LL];
    __shared__ float s_red[8];
    __shared__ float s_beta, s_max, s_inv;

    const int b = blockIdx.x;
    const int tid = threadIdx.x;
    const int lane = tid & 31;
    const int wv = tid >> 5;

    for (int k = tid; k < AA; k += 256) {
        s_h[k] = hsum[b * AA + k];
        s_v[k] = vvec[k];
    }

    // beta gate: block-wide dot(h[b,:], beta_W)
    float bp = 0.f;
    for (int k = tid; k < HH; k += 256) bp += h[b * HH + k] * betaW[k];
    bp = warpReduceSum(bp);
    if (lane == 0) s_red[wv] = bp;
    __syncthreads();
    if (tid == 0) {
        float s = 0.f;
        for (int i = 0; i < 8; ++i) s += s_red[i];
        s += betab[0];
        s_beta = 1.f / (1.f + __expf(-s));
    }
    __syncthreads();

    // e[l]: one wave per l, lanes stride over A (coalesced wa reads)
    for (int l = wv; l < LL; l += 8) {
        const float* wrow = wa + ((long)b * LL + l) * AA;
        float s = 0.f;
        for (int k = lane; k < AA; k += 32)
            s += s_v[k] * tanhf(wrow[k] + s_h[k]);
        s = warpReduceSum(s);
        if (lane == 0) s_e[l] = s;
    }
    __syncthreads();

    // softmax over L=196
    float m = -1e30f;
    for (int l = tid; l < LL; l += 256) m = fmaxf(m, s_e[l]);
    m = warpReduceMax(m);
    if (lane == 0) s_red[wv] = m;
    __syncthreads();
    if (tid == 0) {
        float mm = -1e30f;
        for (int i = 0; i < 8; ++i) mm = fmaxf(mm, s_red[i]);
        s_max = mm;
    }
    __syncthreads();

    float psum = 0.f;
    for (int l = tid; l < LL; l += 256) {
        float ex = __expf(s_e[l] - s_max);
        s_e[l] = ex;
        psum += ex;
    }
    psum = warpReduceSum(psum);
    if (lane == 0) s_red[wv] = psum;
    __syncthreads();
    if (tid == 0) {
        float s = 0.f;
        for (int i = 0; i < 8; ++i) s += s_red[i];
        s_inv = 1.f / s;
    }
    __syncthreads();

    for (int l = tid; l < LL; l += 256) {
        float al = s_e[l] * s_inv;
        s_e[l] = al;
        alphas[((long)b * (TT - 1) + t) * LL + l] = al;
    }
    __syncthreads();

    // ctx[d] = beta * sum_l alpha[l]*a[b,l,d]   (coalesced across threads)
    const float bt = s_beta;
    for (int d = tid; d < DD; d += 256) {
        const float* ab = a + (long)b * LL * DD + d;
        float s = 0.f;
        for (int l = 0; l < LL; ++l) s += s_e[l] * ab[(long)l * DD];
        ctx[b * DD + d] = bt * s;
    }
}

// ---------------------------------------------------------------------------
// LSTM pointwise. gates = ctx@W_ih[E:] + h@W_hh + b_hh (from GEMMs),
// gatesX = emb@W_ih[:E] + b_ih (precomputed, time-major slice at t).
// Updates h, c in place (gates GEMMs for this step already consumed h).
// ---------------------------------------------------------------------------
__device__ __forceinline__ float sigmf(float x) { return 1.f / (1.f + __expf(-x)); }

__global__ __launch_bounds__(256) void lstm_pointwise_kernel(
    const float* __restrict__ gates, const float* __restrict__ gatesX_t,
    float* __restrict__ h, float* __restrict__ c)
{
    int idx = blockIdx.x * 256 + threadIdx.x;   // b*H + j
    if (idx >= BB * HH) return;
    int b = idx >> 9, j = idx & (HH - 1);
    long base = (long)b * G4;
    float gi = gates[base + j]            + gatesX_t[base + j];
    float gf = gates[base + HH + j]       + gatesX_t[base + HH + j];
    float gg = gates[base + 2 * HH + j]   + gatesX_t[base + 2 * HH + j];
    float go = gates[base + 3 * HH + j]   + gatesX_t[base + 3 * HH + j];
    float cn = sigmf(gf) * c[idx] + sigmf(gi) * tanhf(gg);
    float hn = sigmf(go) * tanhf(cn);
    c[idx] = cn;
    h[idx] = hn;
}

// ---------------------------------------------------------------------------
static inline void gemm(hipStream_t s, const float* A, const float* B, float* C,
                        const float* bias, int M, int N, int K,
                        int lda, int ldb, int ldc, int acc, int act)
{
    dim3 grid((N + 63) / 64, (M + 63) / 64);
    gemm_wmma_f32<<<grid, 256, 0, s>>>(A, B, C, bias, M, N, K, lda, ldb, ldc, acc, act);
}

extern "C" void kernel_launch(void* const* d_in, const int* in_sizes, int n_in,
                              void* d_out, int out_size, void* d_ws, size_t ws_size,
                              hipStream_t stream)
{
    const float* a        = (const float*)d_in[0];
    const int*   captions = (const int*)  d_in[1];
    /* lengths d_in[2] unused: all == T */
    const float* embed_W  = (const float*)d_in[3];
    const float* Wa       = (const float*)d_in[4];
    const float* Wh       = (const float*)d_in[5];
    const float* Wc       = (const float*)d_in[6];
    const float* vvec     = (const float*)d_in[7];
    const float* beta_W   = (const float*)d_in[8];
    const float* beta_b   = (const float*)d_in[9];
    const float* W_ih     = (const float*)d_in[10];
    const float* W_hh     = (const float*)d_in[11];
    const float* b_ih     = (const float*)d_in[12];
    const float* b_hh     = (const float*)d_in[13];
    const float* fc_W     = (const float*)d_in[14];
    const float* fc_b     = (const float*)d_in[15];
    const float* initW_h  = (const float*)d_in[16];
    const float* initb_h  = (const float*)d_in[17];
    const float* initW_c  = (const float*)d_in[18];
    const float* initb_c  = (const float*)d_in[19];

    // workspace layout (floats)
    float* ws = (float*)d_ws;
    float* w_mean  = ws;                         // B*D
    float* w_h     = w_mean + BB * DD;           // B*H
    float* w_c     = w_h + BB * HH;              // B*H
    float* w_wa    = w_c + BB * HH;              // B*L*A
    float* w_Whc   = w_wa + (long)BB * LL * AA;  // H*A
    float* w_hsum  = w_Whc + HH * AA;            // B*A
    float* w_ctx   = w_hsum + BB * AA;           // B*D
    float* w_gates = w_ctx + BB * DD;            // B*4H
    float* w_embX  = w_gates + BB * G4;          // (T-1)*B*E
    float* w_gX    = w_embX + (long)(TT - 1) * BB * EE; // (T-1)*B*4H

    float* out_logits = (float*)d_out;                            // ((T-1)*B, V)
    float* out_alphas = out_logits + (long)(TT - 1) * BB * VV;    // (B, T-1, L)

    // ---- one-time setup -------------------------------------------------
    mean_ctx_kernel<<<(BB * DD + 255) / 256, 256, 0, stream>>>(a, w_mean);
    // h0 = tanh(mean@initW_h + b), c0 = tanh(mean@initW_c + b)
    gemm(stream, w_mean, initW_h, w_h, initb_h, BB, HH, DD, DD, HH, HH, 0, 1);
    gemm(stream, w_mean, initW_c, w_c, initb_c, BB, HH, DD, DD, HH, HH, 0, 1);
    // wa = a @ Wa   (flatten (B,L) into M)
    gemm(stream, a, Wa, w_wa, nullptr, BB * LL, AA, DD, DD, AA, AA, 0, 0);
    // Whc = Wh + Wc  (prev_h == h every step)
    add_mat_kernel<<<(HH * AA + 255) / 256, 256, 0, stream>>>(Wh, Wc, w_Whc, HH * AA);
    // embedding gather + precompute emb-half of the LSTM gates for all steps
    embed_gather_kernel<<<((TT - 1) * BB * EE + 255) / 256, 256, 0, stream>>>(
        captions, embed_W, w_embX);
    gemm(stream, w_embX, W_ih, w_gX, b_ih, (TT - 1) * BB, G4, EE, EE, G4, G4, 0, 0);

    // ---- recurrence -----------------------------------------------------
    for (int t = 0; t < TT - 1; ++t) {
        // hsum = h @ (Wh + Wc)
        gemm(stream, w_h, w_Whc, w_hsum, nullptr, BB, AA, HH, HH, AA, AA, 0, 0);
        // fused attention: alpha, beta-gated ctx
        attention_kernel<<<BB, 256, 0, stream>>>(
            w_wa, w_hsum, vvec, w_h, beta_W, beta_b, a, w_ctx, out_alphas, t);
        // gates = ctx @ W_ih[E:,:] + b_hh ; gates += h @ W_hh
        gemm(stream, w_ctx, W_ih + (long)EE * G4, w_gates, b_hh,
             BB, G4, DD, DD, G4, G4, 0, 0);
        gemm(stream, w_h, W_hh, w_gates, nullptr, BB, G4, HH, HH, G4, G4, 1, 0);
        // LSTM pointwise (adds precomputed emb gates slice), updates h,c
        lstm_pointwise_kernel<<<(BB * HH + 255) / 256, 256, 0, stream>>>(
            w_gates, w_gX + (long)t * BB * G4, w_h, w_c);
        // logits_t = h_new @ fc_W + fc_b  -> packed time-major rows
        gemm(stream, w_h, fc_W, out_logits + (long)t * BB * VV, fc_b,
             BB, VV, HH, HH, VV, VV, 0, 0);
    }
}